// RnnCausalAttention_36644660969532
// MI455X (gfx1250) — compile-verified
//
#include <hip/hip_runtime.h>
#include <math.h>
#include <stdint.h>

typedef __attribute__((ext_vector_type(2))) float v2f;
typedef __attribute__((ext_vector_type(4))) float v4f;
typedef __attribute__((ext_vector_type(8))) float v8f;

#define T_  256
#define B_  8
#define H_  256
#define E_  256
#define V_  32000
#define K2H 512
#define M_  (T_ * B_)   // 2048

// ---------------------------------------------------------------------------
// Kernel 1: Pre[t,h,b] = sum_e Wih[h,e] * Embed[X[t,b], e]
// ---------------------------------------------------------------------------
__global__ __launch_bounds__(256) void k_embed_proj(
    const int* __restrict__ X, const float* __restrict__ Embed,
    const float* __restrict__ Wih, float* __restrict__ Pre)
{
    const int t = blockIdx.x;
    const int tid = threadIdx.x;
    __shared__ float sx[E_ * B_];   // sx[e*8+b], 8 KB

    #pragma unroll
    for (int b = 0; b < B_; ++b) {
        const int tok = X[t * B_ + b];
        sx[tid * B_ + b] = Embed[(size_t)tok * E_ + tid];  // coalesced in e=tid
    }
    __syncthreads();

    float acc[B_];
    #pragma unroll
    for (int b = 0; b < B_; ++b) acc[b] = 0.f;

    const float* wrow = Wih + (size_t)tid * E_;
    for (int e = 0; e < E_; ++e) {
        const float w = wrow[e];
        #pragma unroll
        for (int b = 0; b < B_; ++b) acc[b] += w * sx[e * B_ + b]; // LDS broadcast
    }
    #pragma unroll
    for (int b = 0; b < B_; ++b)
        Pre[((size_t)t * H_ + tid) * B_ + b] = acc[b];
}

// ---------------------------------------------------------------------------
// Kernel 2: sequential recurrence  h = tanh(Pre[t] + Whh @ h), Hs[t] = h
// Single workgroup; h lives in LDS.
// ---------------------------------------------------------------------------
__global__ __launch_bounds__(256) void k_rnn(
    const float* __restrict__ Pre, const float* __restrict__ Whh,
    float* __restrict__ Hs)
{
    const int tid = threadIdx.x;
    __shared__ float hb[H_ * B_];   // hb[k*8+b], 8 KB
    #pragma unroll
    for (int b = 0; b < B_; ++b) hb[tid * B_ + b] = 0.f;
    __syncthreads();

    const float* wrow = Whh + (size_t)tid * H_;
    for (int t = 0; t < T_; ++t) {
        float acc[B_];
        #pragma unroll
        for (int b = 0; b < B_; ++b)
            acc[b] = Pre[((size_t)t * H_ + tid) * B_ + b];
        for (int k = 0; k < H_; ++k) {
            const float w = wrow[k];
            #pragma unroll
            for (int b = 0; b < B_; ++b) acc[b] += w * hb[k * B_ + b];
        }
        __syncthreads();   // all reads of old h done
        #pragma unroll
        for (int b = 0; b < B_; ++b) {
            const float v = tanhf(acc[b]);
            hb[tid * B_ + b] = v;
            Hs[((size_t)t * H_ + tid) * B_ + b] = v;
        }
        __syncthreads();   // new h visible
    }
}

// ---------------------------------------------------------------------------
// Kernel 3: Q[t]=Wq@Hs[t], K[t]=Wk@Hs[t]; also scatter Hs into Afc[:, 0:256]
// ---------------------------------------------------------------------------
__global__ __launch_bounds__(256) void k_qk(
    const float* __restrict__ Hs, const float* __restrict__ Wq,
    const float* __restrict__ Wk, float* __restrict__ Q,
    float* __restrict__ Kp, float* __restrict__ Afc)
{
    const int t = blockIdx.x;
    const int tid = threadIdx.x;
    __shared__ float sH[H_ * B_];   // sH[e*8+b]
    for (int idx = tid; idx < H_ * B_; idx += 256)
        sH[idx] = Hs[(size_t)t * H_ * B_ + idx];
    __syncthreads();

    float q[B_], k2[B_];
    #pragma unroll
    for (int b = 0; b < B_; ++b) { q[b] = 0.f; k2[b] = 0.f; }

    const float* wq = Wq + (size_t)tid * H_;
    const float* wk = Wk + (size_t)tid * H_;
    for (int e = 0; e < H_; ++e) {
        const float a = wq[e], c = wk[e];
        #pragma unroll
        for (int b = 0; b < B_; ++b) {
            const float hv = sH[e * B_ + b];
            q[b]  += a * hv;
            k2[b] += c * hv;
        }
    }
    #pragma unroll
    for (int b = 0; b < B_; ++b) {
        Q [((size_t)t * H_ + tid) * B_ + b] = q[b];
        Kp[((size_t)t * H_ + tid) * B_ + b] = k2[b];
        // Afc row m=(t*8+b): first 256 cols = Hs[t,:,b]
        Afc[(size_t)(t * B_ + b) * K2H + tid] = sH[tid * B_ + b];
    }
}

// ---------------------------------------------------------------------------
// Kernel 4: causal Bahdanau attention. One wave per b. Writes Afc[:, 256:512].
// ---------------------------------------------------------------------------
__global__ __launch_bounds__(256) void k_attn(
    const float* __restrict__ Hs, const float* __restrict__ Q,
    const float* __restrict__ Kp, const float* __restrict__ VAtt,
    float* __restrict__ Afc)
{
    const int t    = blockIdx.x;
    const int tid  = threadIdx.x;
    const int lane = tid & 31;
    const int b    = tid >> 5;           // wave id == batch index (8 waves)
    __shared__ float sc[B_][T_];         // scores / alphas, 8 KB

    float q[8], va[8];
    #pragma unroll
    for (int i = 0; i < 8; ++i) {
        const int h = lane + 32 * i;
        q[i]  = Q[((size_t)t * H_ + h) * B_ + b];
        va[i] = VAtt[h];
    }

    // scores s[j] = VAtt . tanh(Q[t] + K[j]),  j <= t
    for (int j = 0; j <= t; ++j) {
        float s = 0.f;
        #pragma unroll
        for (int i = 0; i < 8; ++i) {
            const int h = lane + 32 * i;
            const float kv = Kp[((size_t)j * H_ + h) * B_ + b];
            s += va[i] * tanhf(q[i] + kv);
        }
        #pragma unroll
        for (int off = 16; off > 0; off >>= 1) s += __shfl_xor(s, off, 32);
        if (lane == 0) sc[b][j] = s;
    }
    __syncthreads();

    // softmax over j in [0, t]
    float mx = -INFINITY;
    for (int j = lane; j <= t; j += 32) mx = fmaxf(mx, sc[b][j]);
    #pragma unroll
    for (int off = 16; off > 0; off >>= 1) mx = fmaxf(mx, __shfl_xor(mx, off, 32));
    float sum = 0.f;
    for (int j = lane; j <= t; j += 32) {
        const float e = __expf(sc[b][j] - mx);
        sc[b][j] = e;
        sum += e;
    }
    #pragma unroll
    for (int off = 16; off > 0; off >>= 1) sum += __shfl_xor(sum, off, 32);
    __syncthreads();
    const float inv = 1.f / sum;

    // context Ct[t,h,b] = sum_j alpha_j * Hs[j,h,b]
    float ct[8];
    #pragma unroll
    for (int i = 0; i < 8; ++i) ct[i] = 0.f;
    for (int j = 0; j <= t; ++j) {
        const float al = sc[b][j] * inv;
        #pragma unroll
        for (int i = 0; i < 8; ++i)
            ct[i] += al * Hs[((size_t)j * H_ + (lane + 32 * i)) * B_ + b];
    }
    const int m = t * B_ + b;
    #pragma unroll
    for (int i = 0; i < 8; ++i)
        Afc[(size_t)m * K2H + H_ + lane + 32 * i] = ct[i];
}

// ---------------------------------------------------------------------------
// Kernel 5: logits[m, v] = Afc[m, :] . Wfc[v, :] + bfc[v]   via f32 WMMA.
// M=2048, N=32000, K=512. Grid = 1000 N-strips of 32 columns (two 16-wide
// B-tiles per workgroup -> full 128B store lines, 4 independent WMMA chains).
// Wfc tiles staged to LDS with GLOBAL_LOAD_ASYNC_TO_LDS_B128 (ASYNCcnt).
// ---------------------------------------------------------------------------
#define LDSB_STRIDE 516   // 512 + 4 pad: conflict-free v2f reads, 16B aligned

__global__ __launch_bounds__(256) void k_logits_wmma(
    const float* __restrict__ Afc, const float* __restrict__ Wfc,
    const float* __restrict__ bfc, float* __restrict__ out)
{
    __shared__ float sB[32 * LDSB_STRIDE];   // two Wfc tiles, [n][k], ~64.5 KB
    const int n0  = blockIdx.x * 32;
    const int tid = threadIdx.x;

    // Stage 32 rows of Wfc (each 512 floats) asynchronously into LDS:
    // 16B per lane per op, tracked by ASYNCcnt.
    for (int idx = tid; idx < 32 * 128; idx += 256) {
        const int n  = idx >> 7;
        const int kq = (idx & 127) << 2;
        const unsigned lds_off =
            (unsigned)(uintptr_t)(&sB[n * LDSB_STRIDE + kq]);
        const unsigned long long gaddr =
            (unsigned long long)(uintptr_t)(Wfc + (size_t)(n0 + n) * K2H + kq);
        asm volatile("global_load_async_to_lds_b128 %0, %1, off"
                     :: "v"(lds_off), "v"(gaddr) : "memory");
    }
    asm volatile("s_wait_asynccnt 0x0" ::: "memory");
    __syncthreads();

    const int lane = tid & 31;
    const int wave = tid >> 5;
    const int ln   = lane & 15;          // N (and A-row) index within tile
    const int koff = (lane >> 4) << 1;   // 0 or 2: K-pair select (ISA f32 layout)
    const float bias0 = bfc[n0 + ln];
    const float bias1 = bfc[n0 + 16 + ln];
    const float* Brow0 = &sB[ln * LDSB_STRIDE + koff];
    const float* Brow1 = &sB[(ln + 16) * LDSB_STRIDE + koff];

    for (int mt = wave; mt < M_ / 16; mt += 8) {
        const float* Arow = Afc + (size_t)(mt * 16 + ln) * K2H + koff;
        v8f c0 = {};   // tile0, even K-blocks
        v8f c1 = {};   // tile0, odd  K-blocks
        v8f c2 = {};   // tile1, even K-blocks
        v8f c3 = {};   // tile1, odd  K-blocks
        #pragma unroll 4
        for (int kk = 0; kk < K2H; kk += 8) {
            const v2f a0  = *(const v2f*)(Arow + kk);
            const v2f a1  = *(const v2f*)(Arow + kk + 4);
            const v2f b00 = *(const v2f*)(Brow0 + kk);
            const v2f b01 = *(const v2f*)(Brow0 + kk + 4);
            const v2f b10 = *(const v2f*)(Brow1 + kk);
            const v2f b11 = *(const v2f*)(Brow1 + kk + 4);
            c0 = __builtin_amdgcn_wmma_f32_16x16x4_f32(
                     false, a0, false, b00, (short)0, c0, false, false);
            c1 = __builtin_amdgcn_wmma_f32_16x16x4_f32(
                     false, a1, false, b01, (short)0, c1, false, false);
            c2 = __builtin_amdgcn_wmma_f32_16x16x4_f32(
                     false, a0, false, b10, (short)0, c2, false, false);
            c3 = __builtin_amdgcn_wmma_f32_16x16x4_f32(
                     false, a1, false, b11, (short)0, c3, false, false);
        }
        // C/D layout: VGPR r -> row r + (lane<16 ? 0 : 8), col = lane%16
        const int rbase = mt * 16 + ((lane >> 4) << 3);
        const int col   = n0 + ln;
        #pragma unroll
        for (int r = 0; r < 8; ++r) {
            out[(size_t)(rbase + r) * V_ + col]      = c0[r] + c1[r] + bias0;
            out[(size_t)(rbase + r) * V_ + col + 16] = c2[r] + c3[r] + bias1;
        }
    }
}

// ---------------------------------------------------------------------------
extern "C" void kernel_launch(void* const* d_in, const int* in_sizes, int n_in,
                              void* d_out, int out_size, void* d_ws, size_t ws_size,
                              hipStream_t stream)
{
    const int*   X     = (const int*)  d_in[0];
    const float* Embed = (const float*)d_in[1];
    const float* Wih   = (const float*)d_in[2];
    const float* Whh   = (const float*)d_in[3];
    const float* Wq    = (const float*)d_in[4];
    const float* Wk    = (const float*)d_in[5];
    const float* VAtt  = (const float*)d_in[6];
    const float* Wfc   = (const float*)d_in[7];
    const float* bfc   = (const float*)d_in[8];
    float* out = (float*)d_out;

    const size_t THB = (size_t)T_ * H_ * B_;   // 524288 floats
    float* ws  = (float*)d_ws;
    float* Pre = ws;
    float* Hs  = ws + THB;
    float* Q   = ws + 2 * THB;
    float* Kp  = ws + 3 * THB;
    float* Afc = ws + 4 * THB;                 // [2048 x 512]

    k_embed_proj <<<T_, 256, 0, stream>>>(X, Embed, Wih, Pre);
    k_rnn        <<<1,  256, 0, stream>>>(Pre, Whh, Hs);
    k_qk         <<<T_, 256, 0, stream>>>(Hs, Wq, Wk, Q, Kp, Afc);
    k_attn       <<<T_, 256, 0, stream>>>(Hs, Q, Kp, VAtt, Afc);
    k_logits_wmma<<<V_ / 32, 256, 0, stream>>>(Afc, Wfc, bfc, out);
}